// TemporalGNN_31722628448359
// MI455X (gfx1250) — compile-verified
//
#include <hip/hip_runtime.h>
#include <math.h>

typedef __attribute__((ext_vector_type(2))) float v2f;
typedef __attribute__((ext_vector_type(8))) float v8f;

#define N_NODES 20000
#define N_EDGES 320000
#define F_IN 4
#define PERIODS 12
#define OUT_CH 256
#define HID 128
#define OUT_DIM 12
#define FT (F_IN * PERIODS) /* 48 */

__device__ __forceinline__ v8f wmma4(v2f a, v2f b, v8f c) {
  // D = A(16x4,f32) * B(4x16,f32) + C(16x16,f32)
  return __builtin_amdgcn_wmma_f32_16x16x4_f32(false, a, false, b, (short)0, c,
                                               false, false);
}

// 1 - sigmoid(x) = sigmoid(-x) = 1/(1+exp(x)); branch-free v_exp + v_rcp.
__device__ __forceinline__ float one_minus_sigmoid(float x) {
  return __builtin_amdgcn_rcpf(1.0f + __expf(x));
}

// tanh(x) = sign(x) * (1-e)/(1+e), e = exp(-2|x|); branch-free.
__device__ __forceinline__ float fast_tanhf(float x) {
  float ax = __builtin_fabsf(x);
  float e = __expf(-2.0f * ax);
  float t = (1.0f - e) * __builtin_amdgcn_rcpf(1.0f + e);
  return __builtin_copysignf(t, x);
}

// ---------------------------------------------------------------------------
// K0: fold Wz@Lzw_top -> Mz (4x256), Wh@Lhw_top -> Mh, biases, softmax(attn)
// ---------------------------------------------------------------------------
__global__ void prep_kernel(const float* __restrict__ att,
                            const float* __restrict__ Wz, const float* __restrict__ bz,
                            const float* __restrict__ Wh, const float* __restrict__ bh,
                            const float* __restrict__ Lzw, const float* __restrict__ Lzb,
                            const float* __restrict__ Lhw, const float* __restrict__ Lhb,
                            float* __restrict__ Mz, float* __restrict__ Mh,
                            float* __restrict__ cz, float* __restrict__ ch,
                            float* __restrict__ probs) {
  int j = threadIdx.x; // 0..255
  for (int f = 0; f < F_IN; ++f) {
    float az = 0.f, ah = 0.f;
    for (int i = 0; i < OUT_CH; ++i) {
      az += Wz[f * OUT_CH + i] * Lzw[i * OUT_CH + j];
      ah += Wh[f * OUT_CH + i] * Lhw[i * OUT_CH + j];
    }
    Mz[f * OUT_CH + j] = az;
    Mh[f * OUT_CH + j] = ah;
  }
  float sz = 0.f, sh = 0.f;
  for (int i = 0; i < OUT_CH; ++i) {
    sz += bz[i] * Lzw[i * OUT_CH + j];
    sh += bh[i] * Lhw[i * OUT_CH + j];
  }
  cz[j] = sz + Lzb[j];
  ch[j] = sh + Lhb[j];
  if (j < PERIODS) {
    float m = att[0];
    for (int t = 1; t < PERIODS; ++t) m = fmaxf(m, att[t]);
    float s = 0.f;
    for (int t = 0; t < PERIODS; ++t) s += __expf(att[t] - m);
    probs[j] = __expf(att[j] - m) / s;
  }
}

// ---------------------------------------------------------------------------
// Degree / normalization
// ---------------------------------------------------------------------------
__global__ void deg_init_kernel(float* __restrict__ deg) {
  int n = blockIdx.x * blockDim.x + threadIdx.x;
  if (n < N_NODES) deg[n] = 1.0f; // self-loop weight
}

__global__ void deg_scatter_kernel(const int* __restrict__ ei,
                                   const float* __restrict__ ea,
                                   float* __restrict__ deg) {
  int e = blockIdx.x * blockDim.x + threadIdx.x;
  if (e < N_EDGES) atomicAdd(&deg[ei[N_EDGES + e]], ea[e]);
}

__global__ void dinv_kernel(float* __restrict__ deg) {
  int n = blockIdx.x * blockDim.x + threadIdx.x;
  if (n < N_NODES) {
    float d = deg[n];
    deg[n] = (d > 0.f) ? rsqrtf(d) : 0.f;
  }
}

__global__ void coeff_kernel(const int* __restrict__ ei,
                             const float* __restrict__ ea,
                             const float* __restrict__ dinv,
                             float* __restrict__ coeff) {
  int e = blockIdx.x * blockDim.x + threadIdx.x;
  if (e < N_EDGES) {
    int s = ei[e];
    int d = ei[N_EDGES + e];
    coeff[e] = dinv[s] * ea[e] * dinv[d];
  }
}

// ---------------------------------------------------------------------------
// SpMM: Y = A_norm @ X   (X = x viewed as N x 48)
// ---------------------------------------------------------------------------
__global__ void y_init_kernel(const float* __restrict__ x,
                              const float* __restrict__ dinv,
                              float* __restrict__ Y) {
  int idx = blockIdx.x * blockDim.x + threadIdx.x;
  if (idx < N_NODES * FT) {
    int n = idx / FT;
    float di = dinv[n];
    Y[idx] = di * di * x[idx]; // self-loop contribution
  }
}

__global__ void y_scatter_kernel(const float* __restrict__ x,
                                 const int* __restrict__ ei,
                                 const float* __restrict__ coeff,
                                 float* __restrict__ Y) {
  long long idx = (long long)blockIdx.x * blockDim.x + threadIdx.x;
  if (idx < (long long)N_EDGES * FT) {
    int e = (int)(idx / FT);
    int k = (int)(idx - (long long)e * FT);
    int s = ei[e];
    int d = ei[N_EDGES + e];
    atomicAdd(&Y[(size_t)d * FT + k], coeff[e] * x[(size_t)s * FT + k]);
  }
}

// ---------------------------------------------------------------------------
// K6: fused temporal GRU-collapse using V_WMMA_F32_16X16X4_F32.
// One wave (32 threads) per 16-node tile.
//   H[n, :] = sum_t p_t * (1 - sigmoid(Y_t @ Mz + cz)) * tanh(Y_t @ Mh + ch)
// ---------------------------------------------------------------------------
__global__ __launch_bounds__(32) void temporal_kernel(
    const float* __restrict__ Y, const float* __restrict__ Mz,
    const float* __restrict__ Mh, const float* __restrict__ cz,
    const float* __restrict__ ch, const float* __restrict__ probs,
    float* __restrict__ Hout /* N x 256 */) {
  int tile = blockIdx.x;       // 0..1249
  int lane = threadIdx.x;      // 0..31
  int mrow = lane & 15;        // M (node within tile) / N (col within tile)
  int khalf = lane >> 4;       // 0: K=0,1   1: K=2,3
  int node0 = tile * 16;

  // Preload A operands for all 12 timesteps: A[M][K] = Y[node0+M][f=K] at time t
  float a0[PERIODS], a1[PERIODS];
  const float* yrow = Y + (size_t)(node0 + mrow) * FT;
#pragma unroll
  for (int t = 0; t < PERIODS; ++t) {
    a0[t] = yrow[(2 * khalf + 0) * PERIODS + t];
    a1[t] = yrow[(2 * khalf + 1) * PERIODS + t];
  }
  float pr[PERIODS];
#pragma unroll
  for (int t = 0; t < PERIODS; ++t) pr[t] = probs[t];

  for (int ct = 0; ct < OUT_CH / 16; ++ct) {
    int col = ct * 16 + mrow;
    // B operand: B[K][N], lane holds rows K = 2*khalf + {0,1} at N = mrow
    v2f bzv, bhv;
    bzv[0] = Mz[(2 * khalf + 0) * OUT_CH + col];
    bzv[1] = Mz[(2 * khalf + 1) * OUT_CH + col];
    bhv[0] = Mh[(2 * khalf + 0) * OUT_CH + col];
    bhv[1] = Mh[(2 * khalf + 1) * OUT_CH + col];
    float czv = cz[col];
    float chv = ch[col];

    v8f hacc;
#pragma unroll
    for (int i = 0; i < 8; ++i) hacc[i] = 0.f;

#pragma unroll
    for (int t = 0; t < PERIODS; ++t) {
      v2f a;
      a[0] = a0[t];
      a[1] = a1[t];
      v8f zc, hc;
#pragma unroll
      for (int i = 0; i < 8; ++i) { zc[i] = czv; hc[i] = chv; }
      zc = wmma4(a, bzv, zc); // Y_t @ Mz + cz
      hc = wmma4(a, bhv, hc); // Y_t @ Mh + ch
#pragma unroll
      for (int i = 0; i < 8; ++i) {
        float g = one_minus_sigmoid(zc[i]);  // (1 - Z)
        float ht = fast_tanhf(hc[i]);        // Htil
        hacc[i] += pr[t] * g * ht;
      }
    }
    // C/D layout: reg i, lane -> (M = i + 8*khalf, N = mrow)
#pragma unroll
    for (int i = 0; i < 8; ++i) {
      int node = node0 + i + 8 * khalf;
      Hout[(size_t)node * OUT_CH + col] = hacc[i];
    }
  }
}

// ---------------------------------------------------------------------------
// K7: MLP head. relu(H) @ W1 + b1 -> relu -> @ W2 + b2. One wave per 16 nodes.
// GEMM1: 16x256 @ 256x128 (64 K-steps of WMMA f32 16x16x4, 8 col tiles)
// GEMM2: 16x128 @ 128x16(pad from 12)  via LDS relayout of h1.
// ---------------------------------------------------------------------------
__global__ __launch_bounds__(32) void mlp_kernel(
    const float* __restrict__ H, const float* __restrict__ W1,
    const float* __restrict__ b1, const float* __restrict__ W2,
    const float* __restrict__ b2, float* __restrict__ out /* N x 12 */) {
  __shared__ float lds[16 * HID]; // h1 tile, row-major 16 x 128
  int tile = blockIdx.x;
  int lane = threadIdx.x;
  int mrow = lane & 15;
  int khalf = lane >> 4;
  int node0 = tile * 16;

  v8f acc[HID / 16];
#pragma unroll
  for (int ct = 0; ct < HID / 16; ++ct) {
    float bv = b1[ct * 16 + mrow];
#pragma unroll
    for (int i = 0; i < 8; ++i) acc[ct][i] = bv;
  }

  const float* hrow = H + (size_t)(node0 + mrow) * OUT_CH;
  for (int ks = 0; ks < OUT_CH / 4; ++ks) {
    int k0 = 4 * ks + 2 * khalf;
    v2f a;
    a[0] = fmaxf(hrow[k0 + 0], 0.f); // relu on the fly
    a[1] = fmaxf(hrow[k0 + 1], 0.f);
#pragma unroll
    for (int ct = 0; ct < HID / 16; ++ct) {
      int col = ct * 16 + mrow;
      v2f b;
      b[0] = W1[(k0 + 0) * HID + col];
      b[1] = W1[(k0 + 1) * HID + col];
      acc[ct] = wmma4(a, b, acc[ct]);
    }
  }

  // h1 = relu(acc) -> LDS row-major
#pragma unroll
  for (int ct = 0; ct < HID / 16; ++ct) {
#pragma unroll
    for (int i = 0; i < 8; ++i) {
      int r = i + 8 * khalf;
      int c = ct * 16 + mrow;
      lds[r * HID + c] = fmaxf(acc[ct][i], 0.f);
    }
  }
  __syncthreads();

  // GEMM2: output cols padded 12 -> 16
  int col2 = mrow;
  v8f o;
  float b2v = (col2 < OUT_DIM) ? b2[col2] : 0.f;
#pragma unroll
  for (int i = 0; i < 8; ++i) o[i] = b2v;

  for (int ks = 0; ks < HID / 4; ++ks) {
    int k0 = 4 * ks + 2 * khalf;
    v2f a, b;
    a[0] = lds[mrow * HID + k0 + 0];
    a[1] = lds[mrow * HID + k0 + 1];
    b[0] = (col2 < OUT_DIM) ? W2[(k0 + 0) * OUT_DIM + col2] : 0.f;
    b[1] = (col2 < OUT_DIM) ? W2[(k0 + 1) * OUT_DIM + col2] : 0.f;
    o = wmma4(a, b, o);
  }

#pragma unroll
  for (int i = 0; i < 8; ++i) {
    int node = node0 + i + 8 * khalf;
    if (col2 < OUT_DIM) out[(size_t)node * OUT_DIM + col2] = o[i];
  }
}

// ---------------------------------------------------------------------------
extern "C" void kernel_launch(void* const* d_in, const int* in_sizes, int n_in,
                              void* d_out, int out_size, void* d_ws,
                              size_t ws_size, hipStream_t stream) {
  (void)in_sizes; (void)n_in; (void)out_size; (void)ws_size;
  const float* x   = (const float*)d_in[0];
  const int*   ei  = (const int*)d_in[1];
  const float* ea  = (const float*)d_in[2];
  const float* att = (const float*)d_in[3];
  const float* Wz  = (const float*)d_in[4];
  const float* bz  = (const float*)d_in[5];
  // d_in[6]=Wr, d_in[7]=br, d_in[12]=Lrw, d_in[13]=Lrb are dead (R * Hzero == 0)
  const float* Wh  = (const float*)d_in[8];
  const float* bh  = (const float*)d_in[9];
  const float* Lzw = (const float*)d_in[10];
  const float* Lzb = (const float*)d_in[11];
  const float* Lhw = (const float*)d_in[14];
  const float* Lhb = (const float*)d_in[15];
  const float* W1  = (const float*)d_in[16];
  const float* b1  = (const float*)d_in[17];
  const float* W2  = (const float*)d_in[18];
  const float* b2  = (const float*)d_in[19];

  float* out  = (float*)d_out;                          // N x 12
  float* Hout = out + (size_t)N_NODES * OUT_DIM;        // N x 256 (out_hidden)

  float* ws    = (float*)d_ws;
  float* deg   = ws;                                    // N        (deg -> dinv)
  float* Y     = deg + N_NODES;                         // N*48
  float* coeff = Y + (size_t)N_NODES * FT;              // E
  float* Mz    = coeff + N_EDGES;                       // 4*256
  float* Mh    = Mz + F_IN * OUT_CH;                    // 4*256
  float* cz    = Mh + F_IN * OUT_CH;                    // 256
  float* ch    = cz + OUT_CH;                           // 256
  float* probs = ch + OUT_CH;                           // 12

  prep_kernel<<<1, 256, 0, stream>>>(att, Wz, bz, Wh, bh, Lzw, Lzb, Lhw, Lhb,
                                     Mz, Mh, cz, ch, probs);
  deg_init_kernel<<<(N_NODES + 255) / 256, 256, 0, stream>>>(deg);
  deg_scatter_kernel<<<(N_EDGES + 255) / 256, 256, 0, stream>>>(ei, ea, deg);
  dinv_kernel<<<(N_NODES + 255) / 256, 256, 0, stream>>>(deg);
  coeff_kernel<<<(N_EDGES + 255) / 256, 256, 0, stream>>>(ei, ea, deg, coeff);
  y_init_kernel<<<(N_NODES * FT + 255) / 256, 256, 0, stream>>>(x, deg, Y);
  {
    long long total = (long long)N_EDGES * FT;
    int blocks = (int)((total + 255) / 256);
    y_scatter_kernel<<<blocks, 256, 0, stream>>>(x, ei, coeff, Y);
  }
  temporal_kernel<<<N_NODES / 16, 32, 0, stream>>>(Y, Mz, Mh, cz, ch, probs, Hout);
  mlp_kernel<<<N_NODES / 16, 32, 0, stream>>>(Hout, W1, b1, W2, b2, out);
}